// SigmaDistance_10625749090597
// MI455X (gfx1250) — compile-verified
//
#include <hip/hip_runtime.h>
#include <math.h>

typedef __attribute__((ext_vector_type(2))) float v2f;
typedef __attribute__((ext_vector_type(8))) float v8f;

#define NPTS   4096
#define NBATCH 8
#define MTOT   (NPTS * 3)      // elements entering the std()

// ---------------------------------------------------------------------------
// zero the 32-float workspace (harness poisons ws with 0xAA)
// ---------------------------------------------------------------------------
__global__ void sigma_zero_ws(float* ws) {
    if (threadIdx.x < 32) ws[threadIdx.x] = 0.0f;
}

// ---------------------------------------------------------------------------
// Main kernel: per (row-tile, dir, batch).  NN search via V_WMMA_F32_16X16X4_F32
// on augmented coordinates:
//   A' row i = (-2ax, -2ay, -2az, 1),  B' col j = (bx, by, bz, |b|^2)
//   score(i,j) = |b_j|^2 - 2 a_i . b_j   (argmin_j == nearest neighbor)
// ---------------------------------------------------------------------------
__global__ __launch_bounds__(256) void sigma_nn_kernel(
    const float* __restrict__ x, const float* __restrict__ y,
    float* __restrict__ ws) {

    __shared__ float4 sB[NPTS];   // 64 KB: (bx,by,bz,|b|^2) for all B points

    const int b   = blockIdx.z;
    const int dir = blockIdx.y;
    const float* Aptr = (dir == 0 ? x : y) + (size_t)b * NPTS * 3;
    const float* Bptr = (dir == 0 ? y : x) + (size_t)b * NPTS * 3;

    // cooperative stage of B' into LDS
    for (int i = threadIdx.x; i < NPTS; i += 256) {
        float bx = Bptr[i * 3 + 0];
        float by = Bptr[i * 3 + 1];
        float bz = Bptr[i * 3 + 2];
        sB[i] = make_float4(bx, by, bz, bx * bx + by * by + bz * bz);
    }
    __syncthreads();

    const int  lane   = threadIdx.x & 31;
    const int  wave   = threadIdx.x >> 5;
    const int  row0   = blockIdx.x * 128 + wave * 16;   // this wave's 16 A rows
    const int  nloc   = lane & 15;
    const bool hi     = (lane >= 16);

    // A' tile in 16x4 f32 WMMA layout:
    //   lanes 0-15 : M=lane,    v0=K0, v1=K1   -> (-2ax, -2ay)
    //   lanes16-31 : M=lane-16, v0=K2, v1=K3   -> (-2az,  1.0)
    const int am = row0 + nloc;
    const float ax = Aptr[am * 3 + 0];
    const float ay = Aptr[am * 3 + 1];
    const float az = Aptr[am * 3 + 2];
    v2f a;
    a.x = hi ? (-2.0f * az) : (-2.0f * ax);
    a.y = hi ? 1.0f         : (-2.0f * ay);

    // running min/argmin: C-matrix VGPR r holds row (r or r+8), col = nloc
    float best[8];
    int   bidx[8];
#pragma unroll
    for (int r = 0; r < 8; ++r) { best[r] = 3.0e38f; bidx[r] = 0; }

    // software-pipelined over B tiles: prefetch next tile while computing
    float4 p = sB[nloc];
#pragma unroll 2
    for (int j0 = 0; j0 < NPTS; j0 += 16) {
        const int jn = (j0 + 16) & (NPTS - 1);      // wraps on last iter (harmless)
        const float4 pn = sB[jn + nloc];            // prefetch next B' tile
        // B' tile in 4x16 layout (mirror of A): N=nloc, halves split K 0,1 / 2,3
        v2f bm;
        bm.x = hi ? p.z : p.x;
        bm.y = hi ? p.w : p.y;
        v8f c = {};
        c = __builtin_amdgcn_wmma_f32_16x16x4_f32(
                /*neg_a=*/false, a, /*neg_b=*/false, bm,
                /*c_mod=*/(short)0, c, /*reuse_a=*/false, /*reuse_b=*/false);
#pragma unroll
        for (int r = 0; r < 8; ++r) {
            const float s = c[r];
            if (s < best[r]) { best[r] = s; bidx[r] = j0 + nloc; }
        }
        p = pn;
    }

    // argmin reduction across the 16 lanes of each half-wave
#pragma unroll
    for (int r = 0; r < 8; ++r) {
#pragma unroll
        for (int mask = 8; mask >= 1; mask >>= 1) {
            const float ov = __shfl_xor(best[r], mask, 16);
            const int   oi = __shfl_xor(bidx[r], mask, 16);
            if (ov < best[r] || (ov == best[r] && oi < bidx[r])) {
                best[r] = ov; bidx[r] = oi;
            }
        }
    }

    // lane 0 owns rows row0+0..7, lane 16 owns rows row0+8..15
    if (nloc == 0) {
        const int mbase = row0 + (hi ? 8 : 0);
        float s = 0.0f, s2 = 0.0f;
#pragma unroll
        for (int r = 0; r < 8; ++r) {
            const int mm = mbase + r;
            const float axx = Aptr[mm * 3 + 0];
            const float ayy = Aptr[mm * 3 + 1];
            const float azz = Aptr[mm * 3 + 2];
            const float4 q = sB[bidx[r]];
            const float dx = axx - q.x;
            const float dy = ayy - q.y;
            const float dz = azz - q.z;
            s  += dx + dy + dz;
            s2 += dx * dx + dy * dy + dz * dz;
        }
        float* w = ws + (b * 2 + dir) * 2;
        atomicAdd(&w[0], s);
        atomicAdd(&w[1], s2);
    }
}

// ---------------------------------------------------------------------------
// finalize: std (ddof=1) per (b,dir), sigma = max(dir0,dir1), mean over batch
// ---------------------------------------------------------------------------
__global__ void sigma_finalize(const float* __restrict__ ws,
                               float* __restrict__ out) {
    if (threadIdx.x == 0 && blockIdx.x == 0) {
        float acc = 0.0f;
        const float M = (float)MTOT;
        for (int b = 0; b < NBATCH; ++b) {
            float stds[2];
            for (int d = 0; d < 2; ++d) {
                const float s  = ws[(b * 2 + d) * 2 + 0];
                const float s2 = ws[(b * 2 + d) * 2 + 1];
                const float var = (s2 - s * s / M) / (M - 1.0f);
                stds[d] = sqrtf(fmaxf(var, 0.0f));
            }
            acc += fmaxf(stds[0], stds[1]);
        }
        out[0] = acc / (float)NBATCH;
    }
}

// ---------------------------------------------------------------------------
extern "C" void kernel_launch(void* const* d_in, const int* in_sizes, int n_in,
                              void* d_out, int out_size, void* d_ws, size_t ws_size,
                              hipStream_t stream) {
    (void)in_sizes; (void)n_in; (void)out_size; (void)ws_size;
    const float* x = (const float*)d_in[0];
    const float* y = (const float*)d_in[1];
    float* ws  = (float*)d_ws;
    float* out = (float*)d_out;

    sigma_zero_ws<<<1, 32, 0, stream>>>(ws);

    dim3 grid(NPTS / 128, 2, NBATCH);   // (row tiles, direction, batch)
    sigma_nn_kernel<<<grid, 256, 0, stream>>>(x, y, ws);

    sigma_finalize<<<1, 1, 0, stream>>>(ws, out);
}